// ValueMemory_3822520893832
// MI455X (gfx1250) — compile-verified
//
#include <hip/hip_runtime.h>
#include <math.h>

// Problem constants (match reference)
#define B_DIM   64
#define M_DIM   8192
#define V_DIM   128
#define EPSV    1e-8f

typedef float v2f __attribute__((ext_vector_type(2)));
typedef float v8f __attribute__((ext_vector_type(8)));

// ---------------------------------------------------------------------------
// Pass 1: sim[b,m] = dot(mem[b,m,:]+eps, v[b,:]+eps) / max(|row|*|v|, eps)
// One wave computes 16 rows via V_WMMA_F32_16X16X4_F32 over K=128 in chunks
// of 4.  A-matrix layout (ISA 7.12.2, 32-bit A 16x4): lanes 0-15 hold rows
// 0-15 with K = 4c+{0,1} in the two A VGPRs; lanes 16-31 hold K = 4c+{2,3}.
// B is the query chunk replicated across all 16 columns, so every column of
// the 16x16 accumulator equals the row dot product.  Row |.|^2 accumulates
// in VALU alongside and is folded with shfl_xor(16) across wave halves.
// ---------------------------------------------------------------------------
__global__ __launch_bounds__(256)
void vm_sim_kernel(const float* __restrict__ memory,
                   const float* __restrict__ v,
                   float* __restrict__ sim)
{
    __shared__ float vq[V_DIM];     // v[b,:] + eps
    __shared__ float s_bnorm2;

    const int tid  = threadIdx.x;
    const int b    = blockIdx.y;

    if (tid < V_DIM)
        vq[tid] = v[(size_t)b * V_DIM + tid] + EPSV;
    __syncthreads();
    if (tid == 0) {                  // deterministic serial reduction (tiny)
        float s = 0.f;
        for (int i = 0; i < V_DIM; ++i) s += vq[i] * vq[i];
        s_bnorm2 = s;
    }
    __syncthreads();
    const float bnorm = sqrtf(s_bnorm2);

    const int wave  = tid >> 5;
    const int lane  = tid & 31;
    const int row0  = blockIdx.x * 128 + wave * 16;   // 8 waves x 16 rows
    const int myrow = lane & 15;
    const int koff  = (lane & 16) >> 3;               // 0 (lanes<16) or 2

    const float* rowp =
        memory + ((size_t)b * M_DIM + row0 + myrow) * V_DIM + koff;

    v8f acc0 = {};
    v8f acc1 = {};
    float norm2 = 0.f;

    #pragma unroll
    for (int c = 0; c < 32; c += 2) {
        // gfx1250 prefetch (global_prefetch_b8) a few chunks ahead
        __builtin_prefetch(rowp + 4 * (c + 8), 0, 1);

        float2 m0 = *(const float2*)(rowp + 4 * c);
        float2 m1 = *(const float2*)(rowp + 4 * (c + 1));
        v2f a0; a0.x = m0.x + EPSV; a0.y = m0.y + EPSV;
        v2f a1; a1.x = m1.x + EPSV; a1.y = m1.y + EPSV;
        norm2 += a0.x * a0.x + a0.y * a0.y + a1.x * a1.x + a1.y * a1.y;

        v2f b0; b0.x = vq[4 * c + koff];       b0.y = vq[4 * c + koff + 1];
        v2f b1; b1.x = vq[4 * (c + 1) + koff]; b1.y = vq[4 * (c + 1) + koff + 1];

        // D = A x B + C   (8 args: neg_a, A, neg_b, B, c_mod, C, reuse_a, reuse_b)
        acc0 = __builtin_amdgcn_wmma_f32_16x16x4_f32(
                   false, a0, false, b0, (short)0, acc0, false, false);
        acc1 = __builtin_amdgcn_wmma_f32_16x16x4_f32(
                   false, a1, false, b1, (short)0, acc1, false, false);
    }
    v8f acc = acc0 + acc1;

    // norm2(row j) = partial(lane j) + partial(lane j+16)
    const float n2full = norm2 + __shfl_xor(norm2, 16, 32);

    // C/D layout: VGPR r: lanes 0-15 -> (M=r, N=lane); lanes 16-31 -> (M=r+8).
    // Lane 0 holds rows 0..7 (N=0), lane 16 holds rows 8..15 (N=0).
    #pragma unroll
    for (int r = 0; r < 8; ++r) {
        // lane 0 needs |row r|, lane 16 needs |row r+8|
        const float nr2   = __shfl(n2full, r + ((lane & 16) >> 1), 32);
        const float denom = fmaxf(sqrtf(nr2) * bnorm, EPSV);
        const float s     = acc[r] / denom;
        if (lane == 0 || lane == 16)
            sim[(size_t)b * M_DIM + row0 + ((lane >> 4) << 3) + r] = s;
    }
}

// ---------------------------------------------------------------------------
// Pass 2: per batch -- exact radix select of the kth largest sim, then a
// deterministic compaction of slots >= kth, then the tiny gathered read:
//   read[b,:] = sum_i w_i * mem[b, idx_i, :] + (sum_i w_i^2) * v[b,:]
// ---------------------------------------------------------------------------
#define CAP 256   // headroom over TOPK=64 for ties

__device__ __forceinline__ unsigned f2k(float f) {
    unsigned u = __float_as_uint(f);
    return (u & 0x80000000u) ? ~u : (u | 0x80000000u);   // order-preserving
}
__device__ __forceinline__ float k2f(unsigned k) {
    unsigned u = (k & 0x80000000u) ? (k ^ 0x80000000u) : ~k;
    return __uint_as_float(u);
}

__global__ __launch_bounds__(256)
void vm_select_read_kernel(const float* __restrict__ memory,
                           const float* __restrict__ v,
                           const float* __restrict__ sim,
                           const int*   __restrict__ topk_p,
                           float* __restrict__ out)
{
    __shared__ unsigned keys[M_DIM];        // 32 KB
    __shared__ unsigned hist[256];
    __shared__ unsigned tcnt[257];
    __shared__ unsigned s_prefix, s_mask, s_k;
    __shared__ int      idxl[CAP];
    __shared__ float    wl[CAP];
    __shared__ float    s_sumw2;

    const int tid = threadIdx.x;
    const int b   = blockIdx.x;
    const float* simb = sim + (size_t)b * M_DIM;

    for (int m = tid; m < M_DIM; m += 256)
        keys[m] = f2k(simb[m]);
    if (tid == 0) {
        s_prefix = 0u; s_mask = 0u;
        s_k = (unsigned)topk_p[0];
    }
    __syncthreads();

    // 4 x 8-bit MSB-first radix select for the kth largest key (exact value)
    for (int shift = 24; shift >= 0; shift -= 8) {
        hist[tid] = 0u;
        __syncthreads();
        const unsigned pfx = s_prefix, msk = s_mask;
        for (int m = tid; m < M_DIM; m += 256) {
            const unsigned kk = keys[m];
            if ((kk & msk) == pfx)
                atomicAdd(&hist[(kk >> shift) & 0xFFu], 1u);   // int atomics: order-independent
        }
        __syncthreads();
        if (tid == 0) {
            unsigned k = s_k, cum = 0u; int sel = 0;
            for (int bin = 255; bin >= 0; --bin) {
                const unsigned h = hist[bin];
                if (cum + h >= k) { sel = bin; break; }
                cum += h;
            }
            s_k      = k - cum;
            s_prefix = pfx | ((unsigned)sel << shift);
            s_mask   = msk | (0xFFu << shift);
        }
        __syncthreads();
    }
    const unsigned kth_key = s_prefix;          // kth largest sim, exact bits

    // Deterministic compaction: thread t owns contiguous rows [32t, 32t+32)
    {
        const int base = tid * (M_DIM / 256);
        unsigned c = 0;
        for (int j = 0; j < M_DIM / 256; ++j)
            if (keys[base + j] >= kth_key) ++c;
        tcnt[tid + 1] = c;
        __syncthreads();
        if (tid == 0) {
            tcnt[0] = 0;
            for (int t = 1; t <= 256; ++t) tcnt[t] += tcnt[t - 1];
        }
        __syncthreads();
        unsigned pos = tcnt[tid];
        for (int j = 0; j < M_DIM / 256; ++j) {
            const int m = base + j;
            if (keys[m] >= kth_key) {
                if (pos < CAP) { idxl[pos] = m; wl[pos] = k2f(keys[m]); }
                ++pos;
            }
        }
        __syncthreads();
    }
    const int n = (tcnt[256] < CAP) ? (int)tcnt[256] : CAP;

    if (tid == 0) {                             // deterministic ordered sum
        float s = 0.f;
        for (int i = 0; i < n; ++i) s += wl[i] * wl[i];
        s_sumw2 = s;
    }
    __syncthreads();
    const float sumw2 = s_sumw2;

    // Gathered read: ~64 rows x 128 cols per batch, fully coalesced per row
    if (tid < V_DIM) {
        const float* memb = memory + (size_t)b * M_DIM * V_DIM;
        float acc = 0.f;
        for (int i = 0; i < n; ++i)
            acc += wl[i] * memb[(size_t)idxl[i] * V_DIM + tid];
        out[b * V_DIM + tid] = acc + sumw2 * v[b * V_DIM + tid];
    }
}

// ---------------------------------------------------------------------------
// Launch: inputs = { memory (B*M*V f32), v (B*V f32), topk (1 int) }
// output = read (B*1*1*V f32).  Workspace: sim buffer, B*M floats = 2 MB.
// ---------------------------------------------------------------------------
extern "C" void kernel_launch(void* const* d_in, const int* in_sizes, int n_in,
                              void* d_out, int out_size, void* d_ws, size_t ws_size,
                              hipStream_t stream)
{
    (void)in_sizes; (void)n_in; (void)out_size; (void)ws_size;
    const float* memory = (const float*)d_in[0];
    const float* v      = (const float*)d_in[1];
    const int*   topk   = (const int*)d_in[2];
    float*       out    = (float*)d_out;
    float*       sim    = (float*)d_ws;          // B*M floats = 2 MB

    dim3 g1(M_DIM / 128, B_DIM);                 // 8 waves x 16 rows = 128 rows/block
    vm_sim_kernel<<<g1, 256, 0, stream>>>(memory, v, sim);

    vm_select_read_kernel<<<dim3(B_DIM), 256, 0, stream>>>(memory, v, sim, topk, out);
}